// SpGAT_23605140258860
// MI455X (gfx1250) — compile-verified
//
#include <hip/hip_runtime.h>
#include <hip/hip_bf16.h>
#include <math.h>

// ---------------------------------------------------------------------------
// SpGAT forward for MI455X (gfx1250, wave32).
//
// Roofline: GEMMs total ~26 GFLOP -> negligible on WMMA; runtime is dominated
// by the edge scatter/gather phase (~5.2 GB of L2-resident random traffic,
// ~660M f32 atomics). So: f32 WMMA (16x16x4) for exact GEMMs, wave-per-edge
// coalesced gather + global_atomic_add_f32 scatter for the attention phase.
//
// Workspace layout (floats), ~377 MB total:
//   [AGG (B*8*N*64)][DEN (B*8*N)][AGG2 (B*N*64)][DEN2 (B*N)]   <- zeroed
//   [H1 (B*8*N*64)][AS1][AD1][H2 (B*N*64)][AS2][AD2]
// AGG is reused in place as xc (concat of ELU'd heads, head-major K blocks).
// ---------------------------------------------------------------------------

typedef float v2f __attribute__((ext_vector_type(2)));
typedef float v8f __attribute__((ext_vector_type(8)));

#define GAT_ALPHA   0.2f
#define OUT_SLOPE   0.01f
#define GAT_EPS     1e-16f

static __device__ __forceinline__ float warp_sum32(float v) {
  #pragma unroll
  for (int off = 16; off > 0; off >>= 1) v += __shfl_xor(v, off, 32);
  return v;
}

// ---------------------------------------------------------------------------
__global__ void zero_f32(float* __restrict__ p, long long n) {
  long long i = (long long)blockIdx.x * blockDim.x + threadIdx.x;
  const long long stride = (long long)gridDim.x * blockDim.x;
  for (; i < n; i += stride) p[i] = 0.0f;
}

// ---------------------------------------------------------------------------
// Layer-1 GEMM: H1[b*8+head][n][0:64] = x[b] (N x 256) @ W_heads[head] (256 x 64)
// Block = 128 threads (4 waves). One 16-row M tile per block; wave w computes
// output columns [16w, 16w+16) with V_WMMA_F32_16X16X4_F32, K looped 256/4.
// A tile staged in LDS (row stride 260: 16B-aligned rows, conflict-free cols).
__global__ __launch_bounds__(128) void gat_gemm1(const float* __restrict__ x,
                                                 const float* __restrict__ W,
                                                 float* __restrict__ H1, int N) {
  __shared__ float As[16][260];
  const int bh = blockIdx.y, b = bh >> 3, head = bh & 7;
  const int m0 = blockIdx.x * 16;
  const int tid = threadIdx.x;

  const float* xb = x + ((size_t)b * N + m0) * 256;
  for (int i = tid; i < 16 * 64; i += 128) {
    const int row = i >> 6, c4 = (i & 63) * 4;
    const float4 v = *(const float4*)(xb + row * 256 + c4);
    *(float4*)&As[row][c4] = v;
  }
  __syncthreads();

  const int lane = tid & 31, half = lane >> 4, r = lane & 15;
  const int n0 = (tid >> 5) * 16;
  const float* Wh = W + (size_t)head * 256 * 64;

  v8f c = {};
  #pragma unroll 4
  for (int k = 0; k < 256; k += 4) {
    v2f a, bb;
    // A frag: lanes 0-15 hold K=k,k+1 of row r; lanes 16-31 hold K=k+2,k+3.
    a[0] = As[r][k + 2 * half];
    a[1] = As[r][k + 2 * half + 1];
    // B frag (assumed mirror layout): lane = column n0+r, half selects K pair.
    bb[0] = Wh[(size_t)(k + 2 * half) * 64 + n0 + r];
    bb[1] = Wh[(size_t)(k + 2 * half + 1) * 64 + n0 + r];
    c = __builtin_amdgcn_wmma_f32_16x16x4_f32(false, a, false, bb, (short)0, c,
                                              false, false);
  }
  // C/D layout: lane holds D[v + 8*half][r] in c[v].
  float* out = H1 + ((size_t)bh * N + m0) * 64 + n0 + r;
  #pragma unroll
  for (int v = 0; v < 8; ++v) out[(size_t)(v + 8 * half) * 64] = c[v];
}

// ---------------------------------------------------------------------------
// Layer-2 GEMM: H2[b][n][0:64] = xc[b] (N x 512) @ W_out (512 x 64), where
// xc is the in-place ELU'd AGG buffer with layout [b][head][n][64]; K block
// k maps to (head = k>>6, f = k&63).
__global__ __launch_bounds__(128) void gat_gemm2(const float* __restrict__ XC,
                                                 const float* __restrict__ Wout,
                                                 float* __restrict__ H2, int N) {
  __shared__ float As[16][516];
  const int b = blockIdx.y;
  const int m0 = blockIdx.x * 16;
  const int tid = threadIdx.x;

  for (int i = tid; i < 16 * 128; i += 128) {
    const int row = i >> 7, c4 = (i & 127) * 4;
    const int head = c4 >> 6, f = c4 & 63;
    const float4 v = *(const float4*)(XC + (((size_t)(b * 8 + head) * N + m0 + row) * 64 + f));
    *(float4*)&As[row][c4] = v;
  }
  __syncthreads();

  const int lane = tid & 31, half = lane >> 4, r = lane & 15;
  const int n0 = (tid >> 5) * 16;

  v8f c = {};
  #pragma unroll 4
  for (int k = 0; k < 512; k += 4) {
    v2f a, bb;
    a[0] = As[r][k + 2 * half];
    a[1] = As[r][k + 2 * half + 1];
    bb[0] = Wout[(size_t)(k + 2 * half) * 64 + n0 + r];
    bb[1] = Wout[(size_t)(k + 2 * half + 1) * 64 + n0 + r];
    c = __builtin_amdgcn_wmma_f32_16x16x4_f32(false, a, false, bb, (short)0, c,
                                              false, false);
  }
  float* out = H2 + ((size_t)b * N + m0) * 64 + n0 + r;
  #pragma unroll
  for (int v = 0; v < 8; ++v) out[(size_t)(v + 8 * half) * 64] = c[v];
}

// ---------------------------------------------------------------------------
// Per-node attention scalars: as[i] = h[i]·a[0:64], ad[i] = h[i]·a[64:128].
// One wave32 per node-instance; coalesced 128B loads + butterfly reduce.
__global__ void gat_alpha(const float* __restrict__ Hbuf,
                          const float* __restrict__ avec,
                          float* __restrict__ as, float* __restrict__ ad,
                          int nInst, int N, int nA) {
  const int w = (int)(((long long)blockIdx.x * blockDim.x + threadIdx.x) >> 5);
  const int lane = threadIdx.x & 31;
  if (w >= nInst) return;
  const int aIdx = (w / N) % nA;   // layer1: head = (b*8+h)%8 ; layer2: 0
  const float* h = Hbuf + (size_t)w * 64;
  const float* a = avec + (size_t)aIdx * 128;
  const float v1 = h[lane], v2 = h[lane + 32];
  float s = v1 * a[lane]      + v2 * a[lane + 32];
  float d = v1 * a[lane + 64] + v2 * a[lane + 96];
  s = warp_sum32(s);
  d = warp_sum32(d);
  if (lane == 0) { as[w] = s; ad[w] = d; }
}

// ---------------------------------------------------------------------------
// Edge scatter: one wave32 per (instance, edge) task; lane = feature (x2).
// w = exp(-leakyrelu(as[src]+ad[dst])); denom[src]+=w; agg[src]+=w*h[dst].
__global__ void gat_edge(const int* __restrict__ ei,
                         const float* __restrict__ Hbuf,
                         const float* __restrict__ as,
                         const float* __restrict__ ad,
                         float* __restrict__ agg, float* __restrict__ denom,
                         int N, int E, long long nTasks, int instPerB) {
  const int lane = threadIdx.x & 31;
  long long w  = ((long long)blockIdx.x * blockDim.x + threadIdx.x) >> 5;
  const long long nw = ((long long)gridDim.x * blockDim.x) >> 5;
  for (; w < nTasks; w += nw) {
    const int e = (int)(w % E);
    const int inst = (int)(w / E);          // b*instPerB + h
    const int b = inst / instPerB;
    const int* eb = ei + (size_t)b * 2 * E;
    const int s = eb[e];
    const int d = eb[E + e];
    const size_t ibase = (size_t)inst * N;
    const float z = as[ibase + s] + ad[ibase + d];
    const float wt = __expf(-(z > 0.0f ? z : GAT_ALPHA * z));
    if (lane == 0) atomicAdd(&denom[ibase + s], wt);
    const float* hd = Hbuf + (ibase + d) * 64;
    float* ag = agg + (ibase + s) * 64;
    atomicAdd(&ag[lane],      wt * hd[lane]);
    atomicAdd(&ag[lane + 32], wt * hd[lane + 32]);
  }
}

// ---------------------------------------------------------------------------
// Finalize layer-1 in place: agg = elu(agg / (denom + eps))  (elu alpha = 1).
__global__ void gat_fin_elu(float* __restrict__ agg,
                            const float* __restrict__ denom, long long nElem) {
  long long i = (long long)blockIdx.x * blockDim.x + threadIdx.x;
  const long long stride = (long long)gridDim.x * blockDim.x;
  for (; i < nElem; i += stride) {
    const float v = agg[i] / (denom[i >> 6] + GAT_EPS);
    agg[i] = v > 0.0f ? v : (__expf(v) - 1.0f);
  }
}

// Final output: out = leaky_relu(agg2 / (denom2 + eps), 0.01).
__global__ void gat_fin_out(const float* __restrict__ agg,
                            const float* __restrict__ denom,
                            float* __restrict__ out, long long nElem) {
  long long i = (long long)blockIdx.x * blockDim.x + threadIdx.x;
  const long long stride = (long long)gridDim.x * blockDim.x;
  for (; i < nElem; i += stride) {
    const float v = agg[i] / (denom[i >> 6] + GAT_EPS);
    out[i] = v > 0.0f ? v : OUT_SLOPE * v;
  }
}

// ---------------------------------------------------------------------------
extern "C" void kernel_launch(void* const* d_in, const int* in_sizes, int n_in,
                              void* d_out, int out_size, void* d_ws, size_t ws_size,
                              hipStream_t stream) {
  (void)in_sizes; (void)n_in; (void)out_size; (void)ws_size;
  const float* x       = (const float*)d_in[0];  // [8,10000,256]
  const int*   ei      = (const int*)  d_in[1];  // [8,2,160000]
  const float* W_heads = (const float*)d_in[2];  // [8,256,64]
  const float* a_heads = (const float*)d_in[3];  // [8,128]
  const float* W_out   = (const float*)d_in[4];  // [512,64]
  const float* a_out   = (const float*)d_in[5];  // [128]

  const int B = 8, N = 10000, E = 160000, HEADS = 8;
  const size_t szH1   = (size_t)B * HEADS * N * 64;  // 40.96M floats
  const size_t szDen  = (size_t)B * HEADS * N;       // 640k
  const size_t szAgg2 = (size_t)B * N * 64;          // 5.12M
  const size_t szDen2 = (size_t)B * N;               // 80k

  float* ws   = (float*)d_ws;
  float* AGG  = ws;                 // also xc after ELU finalize
  float* DEN  = AGG  + szH1;
  float* AGG2 = DEN  + szDen;
  float* DEN2 = AGG2 + szAgg2;
  float* H1   = DEN2 + szDen2;
  float* AS1  = H1   + szH1;
  float* AD1  = AS1  + szDen;
  float* H2   = AD1  + szDen;
  float* AS2  = H2   + szAgg2;
  float* AD2  = AS2  + szDen2;

  // 1. zero all accumulators (contiguous range at ws start)
  const long long nz = (long long)(szH1 + szDen + szAgg2 + szDen2);
  zero_f32<<<4096, 256, 0, stream>>>(AGG, nz);

  // 2. layer-1 GEMM (WMMA f32)
  gat_gemm1<<<dim3(N / 16, B * HEADS), 128, 0, stream>>>(x, W_heads, H1, N);

  // 3. per-node attention scalars, layer 1
  gat_alpha<<<(B * HEADS * N + 7) / 8, 256, 0, stream>>>(H1, a_heads, AS1, AD1,
                                                         B * HEADS * N, N, HEADS);
  // 4. edge scatter, layer 1 (10.24M wave-tasks, grid-stride)
  gat_edge<<<16384, 256, 0, stream>>>(ei, H1, AS1, AD1, AGG, DEN, N, E,
                                      (long long)B * HEADS * E, HEADS);
  // 5. ELU finalize in place -> AGG becomes xc
  gat_fin_elu<<<8192, 256, 0, stream>>>(AGG, DEN, (long long)szH1);

  // 6. layer-2 GEMM (WMMA f32, K=512 through head-major xc)
  gat_gemm2<<<dim3(N / 16, B), 128, 0, stream>>>(AGG, W_out, H2, N);

  // 7. per-node attention scalars, layer 2
  gat_alpha<<<(B * N + 7) / 8, 256, 0, stream>>>(H2, a_out, AS2, AD2, B * N, N, 1);

  // 8. edge scatter, layer 2
  gat_edge<<<8192, 256, 0, stream>>>(ei, H2, AS2, AD2, AGG2, DEN2, N, E,
                                     (long long)B * E, 1);
  // 9. final leaky_relu output
  gat_fin_out<<<4096, 256, 0, stream>>>(AGG2, DEN2, (float*)d_out,
                                        (long long)szAgg2);
}